// V19AlgebraUniversalModelA_ActionZ_38233798869652
// MI455X (gfx1250) — compile-verified
//
#include <hip/hip_runtime.h>
#include <math.h>

#define BATCH   256
#define NVIEW   32
#define NSTATES 8192
#define OBS     64
#define YC      32
#define HID     128
#define TSTEPS  8
#define NROWS   33      // NVIEW + 1 (stop row)
#define NROWS_PAD 48    // 3 WMMA row tiles

typedef float v2f __attribute__((ext_vector_type(2)));
typedef float v8f __attribute__((ext_vector_type(8)));

__device__ __forceinline__ float gelu_exact(float x) {
    return 0.5f * x * (1.0f + erff(x * 0.70710678118654752440f));
}

__global__ __launch_bounds__(256, 1)
void oracle_scan_kernel(const int* __restrict__ tables,
                        const int* __restrict__ sigma,
                        const int* __restrict__ xsel,
                        const int* __restrict__ base_obs,
                        const float* __restrict__ W1,
                        const float* __restrict__ b1,
                        const float* __restrict__ W2,
                        const float* __restrict__ b2,
                        float* __restrict__ out_actions,
                        float* __restrict__ out_resp,
                        float* __restrict__ out_ylog) {
    __shared__ unsigned short candA[NSTATES];
    __shared__ unsigned short candB[NSTATES];
    __shared__ unsigned int   counts[NVIEW * OBS];
    __shared__ float hLds[NROWS][HID];
    __shared__ float featLds[NROWS_PAD][4];
    __shared__ float w1s[4][HID];     // K padded 3 -> 4 (row 3 zero)
    __shared__ float b1s[HID];
    __shared__ float w2s[HID];
    __shared__ float expLds[NVIEW];
    __shared__ float zlog[NROWS];
    __shared__ float csig[YC];
    __shared__ unsigned int sK, sNewK, sBuf, sUsed, sStopped, sAc, sR, sIsStop;

    const int tid = threadIdx.x;
    const int b   = blockIdx.x;
    const size_t bOff = (size_t)b * NVIEW * NSTATES;
    const int xb = xsel[b];
    const int bo = base_obs[b];
    const float b2v = b2[0];

    // preload MLP weights into LDS (K-padded)
    for (int i = tid; i < 4 * HID; i += 256)
        w1s[i >> 7][i & 127] = (i < 3 * HID) ? W1[i] : 0.0f;
    for (int i = tid; i < HID; i += 256) { b1s[i] = b1[i]; w2s[i] = W2[i]; }
    if (tid == 0) { sK = 0; sBuf = 0; sUsed = 0; sStopped = 0; }
    __builtin_prefetch(sigma + (size_t)b * NSTATES, 0, 0);  // global_prefetch_b8
    __syncthreads();

    // initial candidate compaction from view 0 (contiguous 32KB scan)
    for (int n = tid; n < NSTATES; n += 256) {
        if (tables[bOff + n] == bo) {
            unsigned int p = atomicAdd(&sK, 1u);
            candA[p] = (unsigned short)n;
        }
    }
    __syncthreads();

    for (int t = 0; t < TSTEPS; ++t) {
        if (sStopped) {  // uniform: stopped -> a = V, r = 0, mask unchanged
            if (tid == 0) {
                out_actions[b * TSTEPS + t] = (float)NVIEW;
                out_resp[b * TSTEPS + t]    = 0.0f;
                sUsed |= (1u << (NVIEW - 1));  // reference sets used[clip(V)=31]
            }
            __syncthreads();
            continue;
        }
        const unsigned int K = sK;
        const unsigned short* cand = sBuf ? candB : candA;
        unsigned short* candNext   = sBuf ? candA : candB;
        const float totalc = (K > 0) ? (float)K : 1.0f;

        // ---- per-view observation histogram over compacted candidates ----
        for (int i = tid; i < NVIEW * OBS; i += 256) counts[i] = 0;
        __syncthreads();
        const unsigned int work = K * NVIEW;
        for (unsigned int i = tid; i < work; i += 256) {
            const unsigned int k = i >> 5;
            const unsigned int v = i & 31;
            const unsigned int n = cand[k];
            const int o = tables[bOff + (size_t)v * NSTATES + n];
            atomicAdd(&counts[v * OBS + o], 1u);
        }
        __syncthreads();

        // ---- exp_size[v] = sum_o counts^2 / total ----
        if (tid < NVIEW) {
            float s = 0.0f;
            for (int o = 0; o < OBS; ++o) {
                const float c = (float)counts[tid * OBS + o];
                s += c * c;
            }
            expLds[tid] = s / totalc;
        }
        __syncthreads();

        // ---- features [48][4], rows >= 33 and feature 3 zero-padded ----
        if (tid < NROWS_PAD) {
            float f0 = 0.f, f1 = 0.f, f2 = 0.f;
            if (tid < NVIEW) {
                const float e = expLds[tid];
                f0 = log2f(fmaxf(e, 1.0f));
                f1 = e - 1.0f;
                f2 = e;
                if (tid == 0) f0 = 1000.0f;      // forbid_view0 oracle cfg
            } else if (tid == NVIEW) {           // stop row
                const bool closed = (totalc <= 1.0f);
                f0 = closed ? 0.0f : 1000.0f;
                f1 = totalc - 1.0f;
                f2 = totalc;
            }
            featLds[tid][0] = fminf(fmaxf(f0, -1000.f), 1000.f);
            featLds[tid][1] = fminf(fmaxf(f1, -1000.f), 1000.f);
            featLds[tid][2] = fminf(fmaxf(f2, -1000.f), 1000.f);
            featLds[tid][3] = 0.0f;
        }
        __syncthreads();

        // ---- MLP layer 1 via V_WMMA_F32_16X16X4_F32: (48x4)@(4x128) ----
        {
            const int wave = tid >> 5;
            const int lane = tid & 31;
            const int half = lane >> 4;        // ISA A16x4 layout: K 0/1 vs 2/3
            const int lidx = lane & 15;
            const int col  = wave * 16 + lidx; // wave owns one hidden col-tile
            v2f bmat;
            bmat[0] = w1s[half * 2 + 0][col];
            bmat[1] = w1s[half * 2 + 1][col];
            for (int rt = 0; rt < 3; ++rt) {
                const int arow = rt * 16 + lidx;
                v2f amat;
                amat[0] = featLds[arow][half * 2 + 0];
                amat[1] = featLds[arow][half * 2 + 1];
                v8f acc = {};
                acc = __builtin_amdgcn_wmma_f32_16x16x4_f32(
                    false, amat, false, bmat, (short)0, acc, false, false);
                for (int r = 0; r < 8; ++r) {
                    const int row = rt * 16 + r + half * 8;  // C/D layout
                    if (row < NROWS)
                        hLds[row][col] = gelu_exact(acc[r] + b1s[col]);
                }
            }
        }
        __syncthreads();

        // ---- layer 2: 33x128 @ 128x1 ----
        if (tid < NROWS) {
            float z = b2v;
            for (int hh = 0; hh < HID; ++hh) z += hLds[tid][hh] * w2s[hh];
            zlog[tid] = z;
        }
        __syncthreads();

        // ---- action selection (thread 0, exact reference semantics) ----
        if (tid == 0) {
            int jstar = 0;
            float best = zlog[0];
            for (int i = 1; i < NROWS; ++i)
                if (zlog[i] > best) { best = zlog[i]; jstar = i; }
            const unsigned int used = sUsed;
            int a;
            const bool jok = (jstar >= 1) &&
                             (jstar == NVIEW || !((used >> jstar) & 1u));
            if (jok) a = jstar;
            else {   // one-hot masked out -> first zero-logit unused view
                a = NVIEW;
                for (int j = 1; j < NVIEW; ++j)
                    if (!((used >> j) & 1u)) { a = j; break; }
            }
            const int isStop = (a == NVIEW);
            const int ac = isStop ? (NVIEW - 1) : a;
            const int r  = isStop ? 0 : tables[bOff + (size_t)ac * NSTATES + xb];
            sUsed = used | (1u << ac);
            sAc = (unsigned int)ac; sR = (unsigned int)r; sIsStop = (unsigned int)isStop;
            if (isStop) sStopped = 1;
            out_actions[b * TSTEPS + t] = (float)a;
            out_resp[b * TSTEPS + t]    = (float)r;
            sNewK = 0;
        }
        __syncthreads();

        // ---- filter candidate list (skip on stop; mask unchanged) ----
        if (!sIsStop) {
            const int ac = (int)sAc;
            const int rr = (int)sR;
            for (unsigned int i = tid; i < K; i += 256) {
                const unsigned int n = cand[i];
                if (tables[bOff + (size_t)ac * NSTATES + n] == rr) {
                    unsigned int p = atomicAdd(&sNewK, 1u);
                    candNext[p] = (unsigned short)n;
                }
            }
            __syncthreads();
            if (tid == 0) { sK = sNewK; sBuf ^= 1; }
        }
        __syncthreads();
    }

    // ---- class posterior: y_logits = log(max(hist(sigma[cand])/K, 1e-9)) ----
    if (tid < YC) csig[tid] = 0.0f;
    __syncthreads();
    {
        const unsigned int K = sK;
        const unsigned short* cand = sBuf ? candB : candA;
        for (unsigned int i = tid; i < K; i += 256) {
            const int n = cand[i];
            atomicAdd(&csig[sigma[(size_t)b * NSTATES + n]], 1.0f);
        }
    }
    __syncthreads();
    if (tid < YC) {
        const float denom = fmaxf((float)sK, 1.0f);
        const float p = csig[tid] / denom;
        out_ylog[b * YC + tid] = logf(fmaxf(p, 1e-9f));
    }
}

extern "C" void kernel_launch(void* const* d_in, const int* in_sizes, int n_in,
                              void* d_out, int out_size, void* d_ws, size_t ws_size,
                              hipStream_t stream) {
    (void)in_sizes; (void)n_in; (void)out_size; (void)d_ws; (void)ws_size;
    const int*   tables   = (const int*)d_in[0];
    const int*   sigma    = (const int*)d_in[1];
    const int*   xsel     = (const int*)d_in[2];
    const int*   base_obs = (const int*)d_in[3];
    const float* W1       = (const float*)d_in[4];
    const float* b1       = (const float*)d_in[5];
    const float* W2       = (const float*)d_in[6];
    const float* b2       = (const float*)d_in[7];
    float* out = (float*)d_out;
    float* out_actions = out;                         // [256,8]
    float* out_resp    = out + BATCH * TSTEPS;        // [256,8]
    float* out_ylog    = out + 2 * BATCH * TSTEPS;    // [256,32]

    oracle_scan_kernel<<<BATCH, 256, 0, stream>>>(
        tables, sigma, xsel, base_obs, W1, b1, W2, b2,
        out_actions, out_resp, out_ylog);
}